// MoD_67482526154878
// MI455X (gfx1250) — compile-verified
//
#include <hip/hip_runtime.h>
#include <math.h>

// ---------------------------------------------------------------------------
// Mixture-of-Depths for MI455X (gfx1250, wave32).
// Only ~12.5% of tokens need the FFN -> gather/compact then run fp32 WMMA
// GEMMs (V_WMMA_F32_16X16X4_F32) on <=1024 rows per batch.
// Round 2: zero-page A rows (no per-iteration EXEC predication) and 32-row
// per-wave M tiles (8 WMMA : 10 VMEM per k-step, half the L2 weight traffic).
// ---------------------------------------------------------------------------

typedef float v2f __attribute__((ext_vector_type(2)));
typedef float v8f __attribute__((ext_vector_type(8)));

#define BATCH   4
#define SEQ     8192
#define DIM     512
#define DFF     2048
#define KTOP    1024      // 0.125 * 8192
#define MAXSEL  1024
#define ZPADN   2048      // zero page covers max K (DFF)

__device__ __forceinline__ float gelu_tanh(float v) {
    // jax.nn.gelu default (approximate=True)
    float t = tanhf(0.7978845608028654f * (v + 0.044715f * v * v * v));
    return 0.5f * v * (1.0f + t);
}

// ---------------- 1) Router: weights[b,s] = dot(x[b,s,:], Wr) + br ----------
__global__ void mod_router(const float* __restrict__ x,
                           const float* __restrict__ Wr,
                           const float* __restrict__ br,
                           float* __restrict__ weights) {
    int wave = threadIdx.x >> 5;
    int lane = threadIdx.x & 31;
    int tok  = blockIdx.x * 8 + wave;
    if (tok >= BATCH * SEQ) return;
    const float* row = x + (size_t)tok * DIM;
    float acc = 0.f;
#pragma unroll
    for (int j = 0; j < DIM / 32; ++j)
        acc += row[j * 32 + lane] * Wr[j * 32 + lane];
#pragma unroll
    for (int off = 16; off; off >>= 1)
        acc += __shfl_xor(acc, off, 32);
    if (lane == 0) weights[tok] = acc + br[0];
}

// ---------------- 2) Exact k-th largest per batch via LDS radix select ------
// Also zeroes this batch's slice of the shared zero page and the counter.
__global__ void mod_topk(const float* __restrict__ weights,
                         float* __restrict__ thr,
                         int* __restrict__ counts,
                         float* __restrict__ zpad) {
    __shared__ unsigned keys[SEQ];        // 32 KB of the 320 KB WGP LDS
    __shared__ unsigned hist[256];
    __shared__ unsigned s_prefix;
    __shared__ unsigned s_rank;
    const int b = blockIdx.x, tid = threadIdx.x;

    // zero page: 4 blocks x 512 floats = 2048
    zpad[b * 512 + tid]       = 0.f;
    zpad[b * 512 + 256 + tid] = 0.f;

    for (int i = tid; i < SEQ; i += blockDim.x) {
        unsigned u = __float_as_uint(weights[b * SEQ + i]);
        // monotonic float->uint transform (ascending unsigned == ascending float)
        u = (u & 0x80000000u) ? ~u : (u | 0x80000000u);
        keys[i] = u;
    }
    if (tid == 0) { s_prefix = 0u; s_rank = KTOP; }
    __syncthreads();

    for (int pass = 0; pass < 4; ++pass) {
        const int shift = 24 - 8 * pass;
        const unsigned maskK = (pass == 0) ? 0u : (0xFFFFFFFFu << (32 - 8 * pass));
        const unsigned pfx = s_prefix;
        if (tid < 256) hist[tid] = 0u;
        __syncthreads();
        for (int i = tid; i < SEQ; i += blockDim.x) {
            unsigned u = keys[i];
            if ((u & maskK) == pfx) atomicAdd(&hist[(u >> shift) & 255u], 1u);
        }
        __syncthreads();
        if (tid == 0) {
            unsigned cum = 0, r = s_rank;
            for (int bin = 255; bin >= 0; --bin) {
                unsigned c = hist[bin];
                if (cum + c >= r) {
                    s_prefix = pfx | ((unsigned)bin << shift);
                    s_rank   = r - cum;
                    break;
                }
                cum += c;
            }
        }
        __syncthreads();
    }
    if (tid == 0) {
        unsigned u = s_prefix;   // key of the k-th largest value
        thr[b] = (u & 0x80000000u) ? __uint_as_float(u ^ 0x80000000u)
                                   : __uint_as_float(~u);
        counts[b] = 0;           // reset compaction counter every call
    }
}

// ---------------- 3a) Passthrough copy out = x (float4) ---------------------
__global__ void mod_copy(const float4* __restrict__ xin,
                         float4* __restrict__ outv, int n4) {
    int i = blockIdx.x * blockDim.x + threadIdx.x;
    if (i < n4) outv[i] = xin[i];
}

// ---------------- 3b) Compact selected token indices ------------------------
__global__ void mod_compact(const float* __restrict__ weights,
                            const float* __restrict__ thr,
                            int* __restrict__ counts,
                            int* __restrict__ sel) {
    int g = blockIdx.x * blockDim.x + threadIdx.x;
    if (g >= BATCH * SEQ) return;
    int b = g >> 13;            // /SEQ
    int s = g & (SEQ - 1);
    if (weights[g] > thr[b]) {  // strict >, matches reference tie handling
        int pos = atomicAdd(&counts[b], 1);   // count <= KTOP-1 always
        sel[b * MAXSEL + pos] = s;
    }
}

// ---------------- 4) H = gelu(gather(x) @ W1 + b1), fp32 WMMA ---------------
// Block: 256 thr = 8 waves; wave -> 32(M) x 64(N) tile; grid (DFF/512, 32, B)
__global__ void mod_ffn1(const float* __restrict__ x,
                         const float* __restrict__ W1,
                         const float* __restrict__ b1,
                         const int* __restrict__ counts,
                         const int* __restrict__ sel,
                         const float* __restrict__ zpad,
                         float* __restrict__ H) {
    const int b    = blockIdx.z;
    const int cnt  = counts[b];
    const int mb   = blockIdx.y * 32;
    if (mb >= cnt) return;
    const int wave = threadIdx.x >> 5;
    const int lane = threadIdx.x & 31;
    const int half = lane >> 4;
    const int lm   = lane & 15;
    const int n0   = blockIdx.x * 512 + wave * 64;

    // A fragment rows: lane lm holds matrix rows mb+lm and mb+16+lm.
    // Invalid rows read the zero page -> no predication in the K loop.
    const int  mi0 = mb + lm,        mi1 = mb + 16 + lm;
    const bool v0  = (mi0 < cnt),    v1  = (mi1 < cnt);
    const int  t0  = v0 ? sel[b * MAXSEL + mi0] : 0;
    const int  t1  = v1 ? sel[b * MAXSEL + mi1] : 0;
    const float* arow0 = v0 ? x + ((size_t)(b * SEQ + t0)) * DIM : zpad;
    const float* arow1 = v1 ? x + ((size_t)(b * SEQ + t1)) * DIM : zpad;

    v8f acc00 = {}, acc01 = {}, acc02 = {}, acc03 = {};
    v8f acc10 = {}, acc11 = {}, acc12 = {}, acc13 = {};

    for (int kk = 0; kk < DIM; kk += 4) {
        const int k0 = kk + 2 * half;
        v2f a0 = *(const v2f*)(arow0 + k0);                   // A[m][k0..k0+1]
        v2f a1 = *(const v2f*)(arow1 + k0);
        const float* bp0 = W1 + (size_t)k0 * DFF + n0 + lm;   // B[k0][n]
        const float* bp1 = bp0 + DFF;                         // B[k0+1][n]
        v2f bf0; bf0[0] = bp0[0];  bf0[1] = bp1[0];
        v2f bf1; bf1[0] = bp0[16]; bf1[1] = bp1[16];
        v2f bf2; bf2[0] = bp0[32]; bf2[1] = bp1[32];
        v2f bf3; bf3[0] = bp0[48]; bf3[1] = bp1[48];
        acc00 = __builtin_amdgcn_wmma_f32_16x16x4_f32(false, a0, false, bf0, (short)0, acc00, false, false);
        acc01 = __builtin_amdgcn_wmma_f32_16x16x4_f32(false, a0, false, bf1, (short)0, acc01, false, false);
        acc02 = __builtin_amdgcn_wmma_f32_16x16x4_f32(false, a0, false, bf2, (short)0, acc02, false, false);
        acc03 = __builtin_amdgcn_wmma_f32_16x16x4_f32(false, a0, false, bf3, (short)0, acc03, false, false);
        acc10 = __builtin_amdgcn_wmma_f32_16x16x4_f32(false, a1, false, bf0, (short)0, acc10, false, false);
        acc11 = __builtin_amdgcn_wmma_f32_16x16x4_f32(false, a1, false, bf1, (short)0, acc11, false, false);
        acc12 = __builtin_amdgcn_wmma_f32_16x16x4_f32(false, a1, false, bf2, (short)0, acc12, false, false);
        acc13 = __builtin_amdgcn_wmma_f32_16x16x4_f32(false, a1, false, bf3, (short)0, acc13, false, false);
    }

    const float bias0 = b1[n0 + lm];
    const float bias1 = b1[n0 + 16 + lm];
    const float bias2 = b1[n0 + 32 + lm];
    const float bias3 = b1[n0 + 48 + lm];
#pragma unroll
    for (int r = 0; r < 8; ++r) {
        int i0 = mb + r + 8 * half;         // D layout: vgpr r, half selects +8
        if (i0 < cnt) {
            float* hrow = H + ((size_t)(b * MAXSEL + i0)) * DFF + n0 + lm;
            hrow[0]  = gelu_tanh(acc00[r] + bias0);
            hrow[16] = gelu_tanh(acc01[r] + bias1);
            hrow[32] = gelu_tanh(acc02[r] + bias2);
            hrow[48] = gelu_tanh(acc03[r] + bias3);
        }
        int i1 = mb + 16 + r + 8 * half;
        if (i1 < cnt) {
            float* hrow = H + ((size_t)(b * MAXSEL + i1)) * DFF + n0 + lm;
            hrow[0]  = gelu_tanh(acc10[r] + bias0);
            hrow[16] = gelu_tanh(acc11[r] + bias1);
            hrow[32] = gelu_tanh(acc12[r] + bias2);
            hrow[48] = gelu_tanh(acc13[r] + bias3);
        }
    }
}

// ---------------- 5) out[sel] = (H @ W2 + b2) * w, scattered ----------------
// Block: 256 thr = 8 waves covering all 512 N cols; 32-row tiles; grid (1,32,B)
__global__ void mod_ffn2(const float* __restrict__ H,
                         const float* __restrict__ W2,
                         const float* __restrict__ b2,
                         const float* __restrict__ weights,
                         const int* __restrict__ counts,
                         const int* __restrict__ sel,
                         const float* __restrict__ zpad,
                         float* __restrict__ out) {
    const int b    = blockIdx.z;
    const int cnt  = counts[b];
    const int mb   = blockIdx.y * 32;
    if (mb >= cnt) return;
    const int wave = threadIdx.x >> 5;
    const int lane = threadIdx.x & 31;
    const int half = lane >> 4;
    const int lm   = lane & 15;
    const int n0   = wave * 64;

    const int  mi0 = mb + lm,     mi1 = mb + 16 + lm;
    const bool v0  = (mi0 < cnt), v1  = (mi1 < cnt);
    const float* arow0 = v0 ? H + ((size_t)(b * MAXSEL + mi0)) * DFF : zpad;
    const float* arow1 = v1 ? H + ((size_t)(b * MAXSEL + mi1)) * DFF : zpad;

    v8f acc00 = {}, acc01 = {}, acc02 = {}, acc03 = {};
    v8f acc10 = {}, acc11 = {}, acc12 = {}, acc13 = {};

    for (int kk = 0; kk < DFF; kk += 4) {
        const int k0 = kk + 2 * half;
        v2f a0 = *(const v2f*)(arow0 + k0);
        v2f a1 = *(const v2f*)(arow1 + k0);
        const float* bp0 = W2 + (size_t)k0 * DIM + n0 + lm;
        const float* bp1 = bp0 + DIM;
        v2f bf0; bf0[0] = bp0[0];  bf0[1] = bp1[0];
        v2f bf1; bf1[0] = bp0[16]; bf1[1] = bp1[16];
        v2f bf2; bf2[0] = bp0[32]; bf2[1] = bp1[32];
        v2f bf3; bf3[0] = bp0[48]; bf3[1] = bp1[48];
        acc00 = __builtin_amdgcn_wmma_f32_16x16x4_f32(false, a0, false, bf0, (short)0, acc00, false, false);
        acc01 = __builtin_amdgcn_wmma_f32_16x16x4_f32(false, a0, false, bf1, (short)0, acc01, false, false);
        acc02 = __builtin_amdgcn_wmma_f32_16x16x4_f32(false, a0, false, bf2, (short)0, acc02, false, false);
        acc03 = __builtin_amdgcn_wmma_f32_16x16x4_f32(false, a0, false, bf3, (short)0, acc03, false, false);
        acc10 = __builtin_amdgcn_wmma_f32_16x16x4_f32(false, a1, false, bf0, (short)0, acc10, false, false);
        acc11 = __builtin_amdgcn_wmma_f32_16x16x4_f32(false, a1, false, bf1, (short)0, acc11, false, false);
        acc12 = __builtin_amdgcn_wmma_f32_16x16x4_f32(false, a1, false, bf2, (short)0, acc12, false, false);
        acc13 = __builtin_amdgcn_wmma_f32_16x16x4_f32(false, a1, false, bf3, (short)0, acc13, false, false);
    }

    const float bias0 = b2[n0 + lm];
    const float bias1 = b2[n0 + 16 + lm];
    const float bias2 = b2[n0 + 32 + lm];
    const float bias3 = b2[n0 + 48 + lm];
#pragma unroll
    for (int r = 0; r < 8; ++r) {
        int i0 = mb + r + 8 * half;
        if (i0 < cnt) {
            int tok  = sel[b * MAXSEL + i0];
            float wv = weights[b * SEQ + tok];
            float* orow = out + ((size_t)(b * SEQ + tok)) * DIM + n0 + lm;
            orow[0]  = (acc00[r] + bias0) * wv;
            orow[16] = (acc01[r] + bias1) * wv;
            orow[32] = (acc02[r] + bias2) * wv;
            orow[48] = (acc03[r] + bias3) * wv;
        }
        int i1 = mb + 16 + r + 8 * half;
        if (i1 < cnt) {
            int tok  = sel[b * MAXSEL + i1];
            float wv = weights[b * SEQ + tok];
            float* orow = out + ((size_t)(b * SEQ + tok)) * DIM + n0 + lm;
            orow[0]  = (acc10[r] + bias0) * wv;
            orow[16] = (acc11[r] + bias1) * wv;
            orow[32] = (acc12[r] + bias2) * wv;
            orow[48] = (acc13[r] + bias3) * wv;
        }
    }
}

// ---------------------------------------------------------------------------
extern "C" void kernel_launch(void* const* d_in, const int* in_sizes, int n_in,
                              void* d_out, int out_size, void* d_ws, size_t ws_size,
                              hipStream_t stream) {
    (void)in_sizes; (void)n_in; (void)out_size; (void)ws_size;
    const float* x  = (const float*)d_in[0];
    // d_in[1] = position_ids (unused by the reference math)
    const float* Wr = (const float*)d_in[2];
    const float* br = (const float*)d_in[3];
    const float* W1 = (const float*)d_in[4];
    const float* b1 = (const float*)d_in[5];
    const float* W2 = (const float*)d_in[6];
    const float* b2 = (const float*)d_in[7];
    float* out = (float*)d_out;

    // Workspace layout (floats): weights | thr | counts | sel | zpad | H
    float* wsf     = (float*)d_ws;
    float* weights = wsf;                                       // 32768
    float* thr     = wsf + BATCH * SEQ;                         // 4
    int*   counts  = (int*)(wsf + BATCH * SEQ + 4);             // 4
    int*   sel     = (int*)(wsf + BATCH * SEQ + 8);             // 4096
    float* zpad    = wsf + BATCH * SEQ + 8 + BATCH * MAXSEL;    // 2048
    float* H       = zpad + ZPADN;                              // 4*1024*2048

    mod_router<<<dim3(BATCH * SEQ / 8), dim3(256), 0, stream>>>(x, Wr, br, weights);
    mod_topk<<<dim3(BATCH), dim3(256), 0, stream>>>(weights, thr, counts, zpad);
    {
        int n4 = BATCH * SEQ * DIM / 4;
        mod_copy<<<dim3((n4 + 255) / 256), dim3(256), 0, stream>>>(
            (const float4*)x, (float4*)out, n4);
    }
    mod_compact<<<dim3(BATCH * SEQ / 256), dim3(256), 0, stream>>>(weights, thr, counts, sel);
    mod_ffn1<<<dim3(DFF / 512, MAXSEL / 32, BATCH), dim3(256), 0, stream>>>(
        x, W1, b1, counts, sel, zpad, H);
    mod_ffn2<<<dim3(1, MAXSEL / 32, BATCH), dim3(256), 0, stream>>>(
        H, W2, b2, weights, counts, sel, zpad, out);
}